// Drifting_74423193305271
// MI455X (gfx1250) — compile-verified
//
#include <hip/hip_runtime.h>
#include <math.h>

// ---------------------------------------------------------------------------
// Problem: B=32, S=512, D=1024, f32 input [B,S,D], bool mask [B,D].
// Algebraic identity:  sum_{t<u} y_t.y_u = 0.5*(||sum_t y_t||^2 - sum_t ||y_t||^2)
// with y_t = (x_t * m) / max(sqrt(max(ss,0)), eps).
// => two bandwidth-bound passes over 67MB (2nd pass L2-resident: 192MB L2).
// Pass 2 column-sum s[b,d] = sum_t y[b,t,d] is done as 1^T * Y on the matrix
// pipe via V_WMMA_F32_16X16X4_F32 (f32 accumulate, cost hidden under BW).
// ---------------------------------------------------------------------------

#define BATCH 32
#define SEQ   512
#define DIM   1024
#define EPSV  1e-8f
#define BETAV 0.1f

typedef __attribute__((ext_vector_type(2))) float v2f;
typedef __attribute__((ext_vector_type(8))) float v8f;

// ws layout (floats): [0]=diag accumulator, [1]=s2 accumulator, [16..16+B*S) inv_n
__global__ void k_init(float* acc) {
    if (threadIdx.x < 2) acc[threadIdx.x] = 0.0f;
}

// Pass 1: one wave32 per (b,t) row. 32 lanes x float4 x 8 iters = 1024 floats.
// Produces inv_n[b*S+t] and accumulates diag term ss*inv^2 (==G_tt/n_t^2).
__global__ void k_row_norms(const float* __restrict__ x,
                            const unsigned char* __restrict__ mask,
                            float* __restrict__ inv_n,
                            float* __restrict__ acc) {
    const int wave = (blockIdx.x * blockDim.x + threadIdx.x) >> 5; // 0..B*S-1 (exact grid)
    const int lane = threadIdx.x & 31;
    const float* row = x + (size_t)wave * DIM;
    const unsigned char* mrow = mask + (size_t)(wave >> 9) * DIM;  // wave/SEQ = b

    float ss = 0.0f;
#pragma unroll
    for (int i = 0; i < 8; ++i) {
        const int d = i * 128 + lane * 4;                 // coalesced 512B/wave/iter
        const float4 v = *(const float4*)(row + d);
        const float a0 = mrow[d + 0] ? v.x : 0.0f;
        const float a1 = mrow[d + 1] ? v.y : 0.0f;
        const float a2 = mrow[d + 2] ? v.z : 0.0f;
        const float a3 = mrow[d + 3] ? v.w : 0.0f;
        ss += a0 * a0 + a1 * a1 + a2 * a2 + a3 * a3;
    }
#pragma unroll
    for (int off = 16; off; off >>= 1) ss += __shfl_xor(ss, off, 32);

    if (lane == 0) {
        float n = sqrtf(fmaxf(ss, 0.0f));
        n = fmaxf(n, EPSV);
        const float inv = 1.0f / n;
        inv_n[wave] = inv;
        atomicAdd(&acc[0], ss * inv * inv);               // sum_t ||y_t||^2
    }
}

// Pass 2: one wave32 per (b, 16-wide d-chunk). Column sums via WMMA f32 16x16x4:
//   D += ones(16x4) * Ytile(4x16)   -> every row of D holds the column sums.
// B-operand lane mapping (32-bit 4x16): VGPR0 lanes0-15 K=0, lanes16-31 K=2;
// VGPR1 K=1 / K=3. With all-ones A the sum over K makes layout exact-proof.
__global__ void k_colsum_wmma(const float* __restrict__ x,
                              const unsigned char* __restrict__ mask,
                              const float* __restrict__ inv_n,
                              float* __restrict__ acc) {
    __shared__ float sInv[SEQ];                           // 2KB of the 320KB/WGP LDS
    const int b  = blockIdx.x >> 3;                       // 32 batches
    const int cg = blockIdx.x & 7;                        // 8 chunk-groups of 8 waves

    for (int i = threadIdx.x; i < SEQ; i += blockDim.x)   // cooperative inv_n stage
        sInv[i] = inv_n[b * SEQ + i];
    __syncthreads();

    const int wave = threadIdx.x >> 5;
    const int lane = threadIdx.x & 31;
    const int d0   = (cg * 8 + wave) * 16;                // this wave's 16 columns
    const int n    = lane & 15;
    const int kb   = (lane < 16) ? 0 : 2;                 // K pair handled by this lane

    const float mf = (float)mask[b * DIM + d0 + n];
    const float* xb = x + (size_t)b * SEQ * DIM + d0 + n;

    const v2f ones = {1.0f, 1.0f};
    v8f c = {};                                           // f32 accumulator tile
    for (int t0 = 0; t0 < SEQ; t0 += 4) {
        const int ta = t0 + kb;
        const int tb = ta + 1;
        const float b0 = xb[(size_t)ta * DIM] * sInv[ta] * mf;
        const float b1 = xb[(size_t)tb * DIM] * sInv[tb] * mf;
        const v2f bv = {b0, b1};
        // 8 args: (neg_a, A, neg_b, B, c_mod, C, reuse_a, reuse_b)
        c = __builtin_amdgcn_wmma_f32_16x16x4_f32(false, ones, false, bv,
                                                  (short)0, c, false, false);
    }

    // D row0: lanes0-15 = s[d0+n]; lanes16-31 duplicate (row M=8) -> 0.5 factor.
    const float s = c[0];
    float v = s * s;
#pragma unroll
    for (int off = 16; off; off >>= 1) v += __shfl_xor(v, off, 32);
    if (lane == 0) atomicAdd(&acc[1], 0.5f * v);          // sum_d s^2 for this chunk
}

__global__ void k_finalize(const float* __restrict__ acc, float* __restrict__ out) {
    const float diag = acc[0];
    const float s2   = acc[1];
    const float total = 0.5f * (s2 - diag);               // strict-upper-tri sum
    const float count = 4186112.0f;                       // 32*512*511/2
    const float avg = total / count;
    out[0] = -logf(1.0f - 0.5f * (avg + 1.0f)) * BETAV;
}

extern "C" void kernel_launch(void* const* d_in, const int* in_sizes, int n_in,
                              void* d_out, int out_size, void* d_ws, size_t ws_size,
                              hipStream_t stream) {
    (void)in_sizes; (void)n_in; (void)out_size; (void)ws_size;
    const float* x            = (const float*)d_in[0];            // [32,512,1024] f32
    const unsigned char* mask = (const unsigned char*)d_in[1];    // [32,1024] bool (1B)
    float* out = (float*)d_out;

    float* acc   = (float*)d_ws;        // [0]=diag, [1]=s2
    float* inv_n = acc + 16;            // B*S floats (64KB)

    k_init<<<1, 32, 0, stream>>>(acc);
    // Pass 1: 16384 rows, 8 waves/block -> 2048 blocks
    k_row_norms<<<2048, 256, 0, stream>>>(x, mask, inv_n, acc);
    // Pass 2: 32 batches * 8 groups -> 256 blocks of 8 waves (one 16-col tile each)
    k_colsum_wmma<<<256, 256, 0, stream>>>(x, mask, inv_n, acc);
    k_finalize<<<1, 1, 0, stream>>>(acc, out);
}